// AtenStftCenter2DWindowPadLeft_53609781789433
// MI455X (gfx1250) — compile-verified
//
#include <hip/hip_runtime.h>
#include <hip/hip_bf16.h>

typedef __attribute__((ext_vector_type(2))) float v2f;
typedef __attribute__((ext_vector_type(8))) float v8f;

#define NFFT   7
#define HOP    2
#define WINLEN 6
#define NFREQ  4

// STFT as GEMM on the CDNA5 matrix core:
//   out[f, col] = sum_n x[2f+n] * Bmat[n, col]
//   Bmat[n, 2k]   =  w[n] * cos(2*pi*n*k/7)
//   Bmat[n, 2k+1] = -w[n] * sin(2*pi*n*k/7)   (rows n=6,7 zero, cols>=8 zero)
// K padded 7 -> 8 = two chained V_WMMA_F32_16X16X4_F32 per 16-frame tile.

__device__ __forceinline__ v8f wmma_f32_k4(v2f a, v2f b, v8f c) {
    return __builtin_amdgcn_wmma_f32_16x16x4_f32(
        /*neg_a=*/false, a, /*neg_b=*/false, b,
        /*c_mod=*/(short)0, c, /*reuse_a=*/false, /*reuse_b=*/false);
}

__global__ __launch_bounds__(256) void stft_wmma_kernel(
    const float* __restrict__ x, const float* __restrict__ w,
    float* __restrict__ out, int L, int F, int fullTiles)
{
    const int lane        = (int)(threadIdx.x & 31);
    const int waveInBlock = (int)(threadIdx.x >> 5);
    const int wavesPerBlk = (int)(blockDim.x >> 5);
    const int waveId      = (int)blockIdx.x * wavesPerBlk + waveInBlock;
    const int numWaves    = (int)gridDim.x * wavesPerBlk;
    const int b           = (int)blockIdx.y;          // batch row: no div/mod

    const int col  = lane & 15;   // N index (output column)
    const int half = lane >> 4;   // 0: lanes 0-15, 1: lanes 16-31
    const int m    = lane & 15;   // A-matrix M (frame within tile)

    // ---- Per-wave B matrix (window folded into DFT basis), built once ----
    // B 4x16 f32: v0 = K(half*2), v1 = K(half*2+1); second WMMA uses K += 4.
    v2f b0, b1;
    {
        const float kf = (float)(col >> 1);
        const float c  = 2.0f * 3.14159265358979323846f / 7.0f;
        float bm[8];
        #pragma unroll
        for (int n = 0; n < 8; ++n) {
            float wv  = (n < WINLEN) ? w[n] : 0.0f;   // w_pad rows 6,7 -> 0
            float ang = c * (float)n * kf;
            float v   = (col & 1) ? (-wv * __sinf(ang)) : (wv * __cosf(ang));
            bm[n] = (col < 2 * NFREQ) ? v : 0.0f;
        }
        b0.x = bm[half * 2 + 0];
        b0.y = bm[half * 2 + 1];
        b1.x = bm[4 + half * 2 + 0];
        b1.y = bm[4 + half * 2 + 1];
    }

    // ---- Per-row / per-lane bases (64-bit math done once) ----
    const float* xrow = x + (size_t)b * (size_t)L;
    const int k = col >> 1;
    // out is complex64 [B, NFREQ, F]: float addr = 2*((b*4+k)*F + f) + (im?1:0)
    float* laneOut = out + ((size_t)b * NFREQ + k) * (size_t)(2 * F) + (col & 1);

    const int      koff     = half * 2;          // K offset for this lane half
    const unsigned clampMax = (unsigned)(L - 2); // even -> keeps b64 alignment
    const unsigned step     = (unsigned)numWaves * 32u; // floats per tile-stride

    // Loop-carried 32-bit load offset (zext -> saddr form) and store pointer.
    unsigned off = (unsigned)(2 * (waveId * 16 + m) + koff);
    float*   p   = laneOut + (size_t)(unsigned)(2 * (waveId * 16 + half * 8));

    // ================= main loop: full 16-frame tiles, no guards ===========
    #pragma unroll 2
    for (int t = waveId; t < fullTiles; t += numWaves) {
        unsigned off4 = off + 4u;
        if (off4 > clampMax) off4 = clampMax;    // only 2nd load can touch row end
        v2f a0 = *(const v2f*)(xrow + off);      // global_load_b64 saddr+voffset
        v2f a1 = *(const v2f*)(xrow + off4);

        v8f acc = {};
        acc = wmma_f32_k4(a0, b0, acc);
        acc = wmma_f32_k4(a1, b1, acc);

        // D vgpr r -> (M = half*8 + r, N = col); lanes col<8 store re/im.
        if (col < 2 * NFREQ) {
            #pragma unroll
            for (int r = 0; r < 8; ++r)
                __builtin_nontemporal_store(acc[r], p + 2 * r);  // imm offsets
        }
        off += step;        // single v_add per iteration
        p   += step;
    }

    // ================= tail tile (F % 16 frames), wave 0 of each row =======
    const int rem = F - (fullTiles << 4);
    if (rem > 0 && waveId == 0) {                // wave-uniform: EXEC stays full
        const int f0 = fullTiles << 4;
        unsigned toff  = (unsigned)(2 * (f0 + m) + koff);
        unsigned toff4 = toff + 4u;
        if (toff  > clampMax) toff  = clampMax;  // OOB frames never stored
        if (toff4 > clampMax) toff4 = clampMax;
        v2f a0 = *(const v2f*)(xrow + toff);
        v2f a1 = *(const v2f*)(xrow + toff4);

        v8f acc = {};
        acc = wmma_f32_k4(a0, b0, acc);
        acc = wmma_f32_k4(a1, b1, acc);

        if (col < 2 * NFREQ) {
            #pragma unroll
            for (int r = 0; r < 8; ++r) {
                const int f = f0 + half * 8 + r;
                if (f < F)
                    __builtin_nontemporal_store(acc[r], laneOut + (size_t)(2 * f));
            }
        }
    }
}

extern "C" void kernel_launch(void* const* d_in, const int* in_sizes, int n_in,
                              void* d_out, int out_size, void* d_ws, size_t ws_size,
                              hipStream_t stream) {
    const float* x = (const float*)d_in[0];
    const float* w = (const float*)d_in[1];
    float* out = (float*)d_out;

    const int Brows = 64;                   // per reference setup
    const int L = in_sizes[0] / Brows;      // 500000
    const int F = 1 + (L - NFFT) / HOP;     // 249997 frames
    const int fullTiles = F >> 4;           // 15624 full 16-frame tiles per row

    dim3 block(256);                        // 8 wave32 waves
    dim3 grid(20, Brows);                   // 160 waves per batch row, ~98 tiles/wave

    hipLaunchKernelGGL(stft_wmma_kernel, grid, block, 0, stream,
                       x, w, out, L, F, fullTiles);
}